// DOAPatternLoss_20109036880302
// MI455X (gfx1250) — compile-verified
//
#include <hip/hip_runtime.h>
#include <hip/hip_bf16.h>

typedef float v2f __attribute__((ext_vector_type(2)));
typedef float v8f __attribute__((ext_vector_type(8)));

#define BLOCK_THREADS 256
#define WAVES_PER_BLOCK (BLOCK_THREADS / 32)
#define GRID_BLOCKS 512

// Init d_out[0] to +inf (as bits) so the atomic-min pass is deterministic.
__global__ void doa_init_kernel(unsigned int* __restrict__ out_bits) {
    if (threadIdx.x == 0 && blockIdx.x == 0) out_bits[0] = 0x7F800000u; // +inf
}

__global__ __launch_bounds__(BLOCK_THREADS)
void doa_min_kernel(const float* __restrict__ P,
                    const float* __restrict__ phases,
                    unsigned int* __restrict__ out_bits,
                    int n_tiles, int rem_rows) {
    const int lane          = threadIdx.x & 31;
    const int wave_in_block = threadIdx.x >> 5;
    const int gwave         = blockIdx.x * WAVES_PER_BLOCK + wave_in_block;
    const int nwaves        = gridDim.x * WAVES_PER_BLOCK;

    // A-operand (16x4 f32) lane mapping: lanes 0-15 -> M=lane, K={0,1};
    // lanes 16-31 -> M=lane-16, K={2,3}. B (4x16) layout matches, so passing
    // the same regs as A and B computes the Gram matrix A*A^T.
    const int mrow    = lane & 15;
    const int colbase = (lane >> 4) << 1; // 0 or 2

    // Preload this lane's phase pair for each of the 8 K-chunks (K step = 4).
    v2f ph[8];
#pragma unroll
    for (int k8 = 0; k8 < 8; ++k8) {
        const float* pp = phases + k8 * 4 + colbase;
        ph[k8].x = pp[0];
        ph[k8].y = pp[1];
    }

    float lmin = __builtin_inff();

    for (int tile = gwave; tile < n_tiles; tile += nwaves) {
        const float* rowp = P + (size_t)(tile * 16 + mrow) * 32;
        v8f acc = {0.f, 0.f, 0.f, 0.f, 0.f, 0.f, 0.f, 0.f};
#pragma unroll
        for (int k8 = 0; k8 < 8; ++k8) {
            const float* q = rowp + k8 * 4 + colbase;
            v2f a;
            a.x = q[0] - ph[k8].x;
            a.y = q[1] - ph[k8].y;
            // D = A * A^T + C : diagonal accumulates per-row sum of squares.
            acc = __builtin_amdgcn_wmma_f32_16x16x4_f32(
                /*neg_a=*/false, a, /*neg_b=*/false, a,
                /*c_mod=*/(short)0, acc, /*reuse_a=*/false, /*reuse_b=*/false);
        }
        // Diagonal extraction from C layout:
        //   rows 0..7  : VGPR i, lane i        (lanes 0-7)
        //   rows 8..15 : VGPR j, lane 24+j     (lanes 24-31)
        const int di = (lane < 8) ? lane : (lane - 24); // negative => no match
        float dval = __builtin_inff();
#pragma unroll
        for (int j = 0; j < 8; ++j)
            if (di == j) dval = acc[j];
        lmin = fminf(lmin, dval);
    }

    // Remainder rows (rows % 16), scalar path; zero for the reference shape.
    {
        const int gtid = blockIdx.x * blockDim.x + threadIdx.x;
        if (gtid < rem_rows) {
            const float* rp = P + (size_t)(n_tiles * 16 + gtid) * 32;
            float s = 0.f;
#pragma unroll
            for (int a = 0; a < 32; ++a) {
                float d = rp[a] - phases[a];
                s = fmaf(d, d, s);
            }
            atomicMin(out_bits, __float_as_uint(s));
        }
    }

    // Wave32 min reduction.
#pragma unroll
    for (int off = 16; off >= 1; off >>= 1)
        lmin = fminf(lmin, __shfl_xor(lmin, off, 32));

    // Block reduction in LDS, one atomic per block.
    __shared__ float smin[WAVES_PER_BLOCK];
    if (lane == 0) smin[wave_in_block] = lmin;
    __syncthreads();
    if (threadIdx.x == 0) {
        float b = smin[0];
#pragma unroll
        for (int w = 1; w < WAVES_PER_BLOCK; ++w) b = fminf(b, smin[w]);
        // All candidate values are >= 0, so uint bit-order == float order.
        atomicMin(out_bits, __float_as_uint(b));
    }
}

extern "C" void kernel_launch(void* const* d_in, const int* in_sizes, int n_in,
                              void* d_out, int out_size, void* d_ws, size_t ws_size,
                              hipStream_t stream) {
    const float* P      = (const float*)d_in[0]; // [R, 32]
    const float* phases = (const float*)d_in[1]; // [32]
    unsigned int* out_bits = (unsigned int*)d_out;

    const int total = in_sizes[0];
    const int A     = (n_in > 1 && in_sizes[1] > 0) ? in_sizes[1] : 32; // 32
    const int rows  = total / A;
    const int n_tiles = rows / 16;
    const int rem     = rows - n_tiles * 16;

    doa_init_kernel<<<1, 32, 0, stream>>>(out_bits);
    doa_min_kernel<<<GRID_BLOCKS, BLOCK_THREADS, 0, stream>>>(
        P, phases, out_bits, n_tiles, rem);
}